// temporal_edge_enhanced_attention_48722109006178
// MI455X (gfx1250) — compile-verified
//
#include <hip/hip_runtime.h>

typedef float v2f __attribute__((ext_vector_type(2)));
typedef float v8f __attribute__((ext_vector_type(8)));

#define B_ 32
#define F_ 64
#define N_ 64
#define C_ 256
#define L_ 16
#define H_ 256   // HID/2

// ---------------------------------------------------------------------------
// One 16x16 tile of G = X(M,C) * W1(H,C)^T using V_WMMA_F32_16X16X4_F32.
// A-matrix f32 16x4 layout (ISA 7.12.2): lanes 0-15 -> M=lane, VGPR0=K0,VGPR1=K1;
// lanes 16-31 -> M=lane-16, VGPR0=K2, VGPR1=K3.  B 4x16 mirrored (n = lane&15).
// ---------------------------------------------------------------------------
__device__ inline v8f wmma_tile_f32(const float* __restrict__ X,
                                    const float* __restrict__ W1,
                                    int row0, int col0) {
  const int lane = threadIdx.x & 31;
  const int mn   = lane & 15;            // M for A, N for B
  const int kb   = (lane >> 4) << 1;     // 0 or 2
  const float* __restrict__ xrow = X  + (size_t)(row0 + mn) * C_;
  const float* __restrict__ wrow = W1 + (size_t)(col0 + mn) * C_;
  v8f acc = {};
#pragma unroll 4
  for (int c = 0; c < C_; c += 4) {
    v2f a = *(const v2f*)(xrow + c + kb);   // K = c+kb, c+kb+1 (8B aligned)
    v2f b = *(const v2f*)(wrow + c + kb);
    acc = __builtin_amdgcn_wmma_f32_16x16x4_f32(
        /*neg_a=*/false, a, /*neg_b=*/false, b,
        /*c_mod=*/(short)0, acc, /*reuse_a=*/false, /*reuse_b=*/false);
  }
  return acc;
}

// ---------------------------------------------------------------------------
// Kernel 1: G[(b*N + n), h] = sum_c src[b,n,c] * W1[h,c]   (2048 x 256, K=256)
// One wave per 16x16 tile; 2048 tiles -> 256 blocks of 256 threads (8 waves).
// ---------------------------------------------------------------------------
__global__ void __launch_bounds__(256)
gemm_g_kernel(const float* __restrict__ src, const float* __restrict__ W1,
              float* __restrict__ G) {
  const int wid  = (blockIdx.x * 256 + threadIdx.x) >> 5;  // 0..2047
  const int tm   = wid >> 4;           // 0..127  (rows of 16)
  const int tn   = wid & 15;           // 0..15   (cols of 16)
  const int row0 = tm << 4, col0 = tn << 4;
  v8f d = wmma_tile_f32(src, W1, row0, col0);
  const int lane = threadIdx.x & 31;
  const int n    = lane & 15;
  const int mb   = (lane >> 4) << 3;   // 0 or 8
  float* gp = G + (size_t)(row0 + mb) * H_ + col0 + n;
#pragma unroll
  for (int r = 0; r < 8; ++r) gp[(size_t)r * H_] = d[r];
}

// ---------------------------------------------------------------------------
// Edge stage (shared by both paths): g = G[b] already in LDS.
// Telescoped: out[b,i,j] = W2 . prelu(g[spd[i,j,len-1]] - g[spd[i,j,0]]).
// Lane l owns channels [4l,4l+4) and [128+4l,128+4l+4): conflict-free b128 LDS.
// ---------------------------------------------------------------------------
__device__ inline void edge_stage(const float* __restrict__ g,
                                  const float* __restrict__ W2,
                                  float pa,
                                  const int* __restrict__ spd,
                                  const int* __restrict__ spd_len,
                                  float* __restrict__ out,
                                  int b, int i) {
  const int tid  = threadIdx.x;
  const int lane = tid & 31;
  const int wv   = tid >> 5;                 // 0..7
  const float4 w2a = *(const float4*)(W2 + lane * 4);
  const float4 w2b = *(const float4*)(W2 + 128 + lane * 4);

  for (int j = wv; j < F_; j += 8) {
    const int idx = i * F_ + j;
    const int len = spd_len[idx];            // in [2,16]
    const int e   = spd[idx * L_ + (len - 1)];
    const int h   = spd[idx * L_];
    const float4* re = (const float4*)(g + (size_t)e * H_) + lane;
    const float4* rh = (const float4*)(g + (size_t)h * H_) + lane;
    const float4 e0 = re[0],  e1 = re[32];
    const float4 q0 = rh[0],  q1 = rh[32];

    float v, s = 0.f;
    v = e0.x - q0.x; s += (v >= 0.f ? v : pa * v) * w2a.x;
    v = e0.y - q0.y; s += (v >= 0.f ? v : pa * v) * w2a.y;
    v = e0.z - q0.z; s += (v >= 0.f ? v : pa * v) * w2a.z;
    v = e0.w - q0.w; s += (v >= 0.f ? v : pa * v) * w2a.w;
    v = e1.x - q1.x; s += (v >= 0.f ? v : pa * v) * w2b.x;
    v = e1.y - q1.y; s += (v >= 0.f ? v : pa * v) * w2b.y;
    v = e1.z - q1.z; s += (v >= 0.f ? v : pa * v) * w2b.z;
    v = e1.w - q1.w; s += (v >= 0.f ? v : pa * v) * w2b.w;

#pragma unroll
    for (int off = 16; off >= 1; off >>= 1) s += __shfl_xor(s, off, 32);
    if (lane == 0) out[((size_t)b * F_ + i) * F_ + j] = s;
  }
}

// ---------------------------------------------------------------------------
// Kernel 2: per (i, b): stage G[b] (64 KB) -> LDS, then 64 edges.
// ---------------------------------------------------------------------------
__global__ void __launch_bounds__(256)
edge_kernel(const float* __restrict__ G, const float* __restrict__ W2,
            const float* __restrict__ prelu_a,
            const int* __restrict__ spd, const int* __restrict__ spd_len,
            float* __restrict__ out) {
  __shared__ __align__(16) float g[N_ * H_];   // 64 KB
  const int i = blockIdx.x, b = blockIdx.y;
  const float4* gs = (const float4*)(G + (size_t)b * N_ * H_);
  float4* gd = (float4*)g;
#pragma unroll 4
  for (int v = threadIdx.x; v < (N_ * H_) / 4; v += 256) gd[v] = gs[v];
  __syncthreads();
  edge_stage(g, W2, prelu_a[0], spd, spd_len, out, b, i);
}

// ---------------------------------------------------------------------------
// Fallback (no scratch): per (i, b) block computes G[b] into LDS with WMMA
// (4x16 tiles of 16x16, 8 waves -> 8 tiles each), then runs the edge stage.
// ---------------------------------------------------------------------------
__global__ void __launch_bounds__(256)
fused_kernel(const float* __restrict__ src, const float* __restrict__ W1,
             const float* __restrict__ W2, const float* __restrict__ prelu_a,
             const int* __restrict__ spd, const int* __restrict__ spd_len,
             float* __restrict__ out) {
  __shared__ __align__(16) float g[N_ * H_];   // 64 KB
  const int i = blockIdx.x, b = blockIdx.y;
  const int lane = threadIdx.x & 31;
  const int wv   = threadIdx.x >> 5;
  const float* Xb = src + (size_t)b * N_ * C_;
  for (int t = wv; t < (N_ / 16) * (H_ / 16); t += 8) {
    const int tm = t >> 4, tn = t & 15;
    v8f d = wmma_tile_f32(Xb, W1, tm << 4, tn << 4);
    const int n  = lane & 15;
    const int mb = (lane >> 4) << 3;
#pragma unroll
    for (int r = 0; r < 8; ++r)
      g[(size_t)((tm << 4) + mb + r) * H_ + (tn << 4) + n] = d[r];
  }
  __syncthreads();
  edge_stage(g, W2, prelu_a[0], spd, spd_len, out, b, i);
}

// ---------------------------------------------------------------------------
extern "C" void kernel_launch(void* const* d_in, const int* in_sizes, int n_in,
                              void* d_out, int out_size, void* d_ws, size_t ws_size,
                              hipStream_t stream) {
  const float* src     = (const float*)d_in[0];   // (B,N,C)
  const float* W1      = (const float*)d_in[1];   // (H,C)
  const float* W2      = (const float*)d_in[2];   // (1,H)
  const float* prelu_a = (const float*)d_in[3];   // scalar
  const int*   spd     = (const int*)d_in[4];     // (F,F,L)
  const int*   spd_len = (const int*)d_in[5];     // (F,F)
  float*       out     = (float*)d_out;           // (B,F,F,1)

  const size_t g_bytes = (size_t)B_ * N_ * H_ * sizeof(float);  // 2 MB
  dim3 egrid(F_, B_);
  if (ws_size >= g_bytes) {
    float* G = (float*)d_ws;
    gemm_g_kernel<<<dim3((B_ * N_ / 16) * (H_ / 16) / 8), 256, 0, stream>>>(src, W1, G);
    edge_kernel<<<egrid, 256, 0, stream>>>(G, W2, prelu_a, spd, spd_len, out);
  } else {
    fused_kernel<<<egrid, 256, 0, stream>>>(src, W1, W2, prelu_a, spd, spd_len, out);
  }
}